// OptimizedFractalDenoise1D_25529285607636
// MI455X (gfx1250) — compile-verified
//
#include <hip/hip_runtime.h>
#include <stdint.h>

typedef __attribute__((ext_vector_type(2))) float        v2f;
typedef __attribute__((ext_vector_type(8))) float        v8f;
typedef __attribute__((ext_vector_type(4))) unsigned int v4u;
typedef __attribute__((ext_vector_type(8))) int          v8i;
typedef __attribute__((ext_vector_type(4))) int          v4i;

#define LROW     65536
#define NROWS    512
#define NTHREADS 1024
#define NWAVES   32
#define NTILES   256       // 256-elem (16x16) tiles per row
#define SPW      8         // tile-strides per wave: tile = wid + 32*s
#define SUBR     8192      // elements per phase-2 sub-round (32 tiles)
#define NSUB     8

__device__ __forceinline__ int refl(int i) {
  // numpy/jax 'reflect' (edge not repeated); offsets stay within [-10, L+10]
  i = (i < 0) ? -i : i;
  i = (i >= LROW) ? (2 * LROW - 2 - i) : i;
  return i;
}

__device__ __forceinline__ v8f wmma4(v2f a, v2f b, v8f c) {
  // D(16x16,f32) = A(16x4,f32) x B(4x16,f32) + C
  return __builtin_amdgcn_wmma_f32_16x16x4_f32(
      /*neg_a=*/false, a, /*neg_b=*/false, b,
      /*c_mod=*/(short)0, c, /*reuse_a=*/false, /*reuse_b=*/false);
}

// Preload 7 A-operand chunks for one tile into distinct registers so ds_loads
// batch ahead of the WMMA chain. MODE 0: plain; 1: reflect edges; 2: offsets
// below tb0 come from the saved left-halo (old values of previous sub-round).
template <int MODE>
__device__ __forceinline__ void loadA(const float* __restrict__ row,
                                      const float* __restrict__ haloL,
                                      int base, int tb0, v2f a[7]) {
#pragma unroll
  for (int c = 0; c < 7; ++c) {
    int o0 = base + 4 * c;
    int o1 = o0 + 1;
    if (MODE == 0) {
      a[c].x = row[o0];
      a[c].y = row[o1];
    } else if (MODE == 1) {
      a[c].x = row[refl(o0)];
      a[c].y = row[refl(o1)];
    } else {
      a[c].x = (o0 < tb0) ? haloL[o0 - tb0 + 5] : row[o0];
      a[c].y = (o1 < tb0) ? haloL[o1 - tb0 + 5] : row[o1];
    }
  }
}

__global__ __launch_bounds__(NTHREADS, 1)
void fractal_denoise_row(const float* __restrict__ x, float* __restrict__ out) {
  __shared__ float row[LROW];    // 256 KB: entire row resident in LDS
  __shared__ float tmp[SUBR];    // 32 KB: phase-2 sub-round output buffer
  __shared__ float haloL[8];     // old left-halo across sub-round boundary
  __shared__ float redS[NWAVES];
  __shared__ float redSS[NWAVES];
  __shared__ float thrS;

  const int lane  = threadIdx.x & 31;
  const int wid   = threadIdx.x >> 5;
  const int n     = lane & 15;        // N column of 16x16 tile
  const int hi    = lane >> 4;        // half-wave select
  const int ks    = hi << 1;          // K lane split for A/B vgpr pair
  const int mbase = n << 4;           // 16*M for A-operand loads (M = lane&15)
  const int vbase = (hi << 7) + n;    // D readout: pos = tb + 16*(v+8*hi) + n
  const int abase = mbase + ks - 5;   // A-chunk base offset within tile

  const int    rowid  = blockIdx.x;
  const size_t rowoff = (size_t)rowid * LROW;

  // ---- TDM: DMA whole row (256 KB) global -> LDS, one descriptor ----
  if (wid == 0) {
    uint64_t ga     = (uint64_t)(uintptr_t)(x + rowoff);
    unsigned ldsoff = (unsigned)(uintptr_t)&row[0];
    // D# group0: count=1 | lds_addr | global_addr[56:0] | type=2
    v4u g0 = {1u, ldsoff, (unsigned)ga,
              (unsigned)((ga >> 32) & 0x01FFFFFFu) | 0x80000000u};
    // D# group1: data_size=4B; tensor 16384x4, tile 16384x4, dim0_stride=16384
    v8i g1 = {0x00020000, 0x40000000, 0x00040000, 0x40000000, 4, 16384, 0, 0};
    v4i gz4 = {0, 0, 0, 0};
    v8i gz8 = {0, 0, 0, 0, 0, 0, 0, 0};
    __builtin_amdgcn_tensor_load_to_lds(g0, g1, gz4, gz4, gz8, 0);
    __builtin_amdgcn_s_wait_tensorcnt(0);
  }
  __syncthreads();

  // Filter (B-matrix) registers: constant for all tiles of this lane.
  // B[k][n] = 1/W if |k-5-n| <= halfwidth else 0; k = 4c + ks (+1 for .y)
  v2f bt[7], bl[7];
#pragma unroll
  for (int c = 0; c < 7; ++c) {
    int k0 = 4 * c + ks;
    int d0 = k0 - 5 - n, d1 = k0 - 4 - n;
    bt[c].x = (d0 >= -5 && d0 <= 5) ? (1.0f / 11.0f) : 0.0f;
    bt[c].y = (d1 >= -5 && d1 <= 5) ? (1.0f / 11.0f) : 0.0f;
    bl[c].x = (d0 >= -2 && d0 <= 2) ? (1.0f / 5.0f) : 0.0f;
    bl[c].y = (d1 >= -2 && d1 <= 2) ? (1.0f / 5.0f) : 0.0f;
  }

  for (int it = 0; it < 2; ++it) {
    // ---- phase 1: local(5-tap) via WMMA; accumulate residual stats ----
    float s = 0.0f, ss = 0.0f;
#pragma unroll 1
    for (int sIdx = 0; sIdx < SPW; ++sIdx) {
      const int  tile = (sIdx << 5) + wid;
      const int  tb   = tile << 8;
      const bool edge = (tile == 0) || (tile == NTILES - 1);  // wave-uniform
      v2f a[7];
      if (edge) loadA<1>(row, haloL, tb + abase, 0, a);
      else      loadA<0>(row, haloL, tb + abase, 0, a);
      v8f acc = {0.f, 0.f, 0.f, 0.f, 0.f, 0.f, 0.f, 0.f};
#pragma unroll
      for (int c = 0; c < 7; ++c) acc = wmma4(a[c], bl[c], acc);
#pragma unroll
      for (int v = 0; v < 8; ++v) {
        float xv = row[tb + vbase + (v << 4)];
        float r  = xv - acc[v];
        s  += r;
        ss += r * r;
      }
    }
#pragma unroll
    for (int o = 16; o > 0; o >>= 1) {
      s  += __shfl_xor(s, o, 32);
      ss += __shfl_xor(ss, o, 32);
    }
    if (lane == 0) { redS[wid] = s; redSS[wid] = ss; }
    __syncthreads();
    if (threadIdx.x < 32) {
      float s2 = redS[lane], ss2 = redSS[lane];
#pragma unroll
      for (int o = 16; o > 0; o >>= 1) {
        s2  += __shfl_xor(s2, o, 32);
        ss2 += __shfl_xor(ss2, o, 32);
      }
      if (lane == 0) {
        const float Nf = (float)LROW;
        float var = (ss2 - s2 * s2 / Nf) / (Nf - 1.0f);  // ddof=1
        var = var > 0.0f ? var : 0.0f;
        float sc = sqrtf(var);
        sc = sc > 1e-6f ? sc : 1e-6f;
        thrS = sc * 3.5f;
      }
    }
    __syncthreads();
    const float thr = thrS;

    // ---- phase 2: trend+local via WMMA (shared A), damp + combine ----
    if (it == 0) {
      // Sub-rounds: one tile per wave per sub-round; results staged in tmp,
      // copied back into row with a 5-element old-value halo save.
#pragma unroll 1
      for (int k = 0; k < NSUB; ++k) {
        const int tile = (k << 5) + wid;
        const int tb   = tile << 8;
        const int tb0  = k * SUBR;
        v2f a[7];
        if (wid == 0 && k == 0)              loadA<1>(row, haloL, tb + abase, tb0, a);
        else if (wid == NWAVES - 1 && k == NSUB - 1)
                                             loadA<1>(row, haloL, tb + abase, tb0, a);
        else if (wid == 0)                   loadA<2>(row, haloL, tb + abase, tb0, a);
        else                                 loadA<0>(row, haloL, tb + abase, tb0, a);
        v8f accl = {0.f, 0.f, 0.f, 0.f, 0.f, 0.f, 0.f, 0.f};
        v8f acct = {0.f, 0.f, 0.f, 0.f, 0.f, 0.f, 0.f, 0.f};
#pragma unroll
        for (int c = 0; c < 7; ++c) {
          accl = wmma4(a[c], bl[c], accl);
          acct = wmma4(a[c], bt[c], acct);
        }
#pragma unroll
        for (int v = 0; v < 8; ++v) {
          int   pos = tb + vbase + (v << 4);
          float xv  = row[pos];
          float l   = accl[v];
          float t   = acct[v];
          float r   = xv - l;
          float rd  = (fabsf(r) > thr) ? r * 0.35f : r;
          tmp[pos - tb0] = 0.4f * l + 0.6f * t + 0.85f * rd;
        }
        __syncthreads();  // all tiles of this sub-round staged in tmp
        const int tail = tb0 + SUBR - 5;  // halo needed by next sub-round
#pragma unroll
        for (int e = 0; e < SUBR / NTHREADS; ++e) {
          int   p  = tb0 + (int)threadIdx.x + (e << 10);
          float nv = tmp[p - tb0];
          if (p >= tail && k < NSUB - 1) haloL[p - tail] = row[p];  // save old
          row[p] = nv;
        }
        __syncthreads();  // row range updated; halo saved
      }
    } else {
      // Final iteration: stream results straight to HBM.
#pragma unroll 1
      for (int sIdx = 0; sIdx < SPW; ++sIdx) {
        const int  tile = (sIdx << 5) + wid;
        const int  tb   = tile << 8;
        const bool edge = (tile == 0) || (tile == NTILES - 1);
        v2f a[7];
        if (edge) loadA<1>(row, haloL, tb + abase, 0, a);
        else      loadA<0>(row, haloL, tb + abase, 0, a);
        v8f accl = {0.f, 0.f, 0.f, 0.f, 0.f, 0.f, 0.f, 0.f};
        v8f acct = {0.f, 0.f, 0.f, 0.f, 0.f, 0.f, 0.f, 0.f};
#pragma unroll
        for (int c = 0; c < 7; ++c) {
          accl = wmma4(a[c], bl[c], accl);
          acct = wmma4(a[c], bt[c], acct);
        }
#pragma unroll
        for (int v = 0; v < 8; ++v) {
          int   pos = tb + vbase + (v << 4);
          float xv  = row[pos];
          float l   = accl[v];
          float t   = acct[v];
          float r   = xv - l;
          float rd  = (fabsf(r) > thr) ? r * 0.35f : r;
          out[rowoff + pos] = 0.4f * l + 0.6f * t + 0.85f * rd;
        }
      }
    }
  }
}

extern "C" void kernel_launch(void* const* d_in, const int* in_sizes, int n_in,
                              void* d_out, int out_size, void* d_ws, size_t ws_size,
                              hipStream_t stream) {
  (void)in_sizes; (void)n_in; (void)d_ws; (void)ws_size; (void)out_size;
  const float* x   = (const float*)d_in[0];   // [16,32,65536] f32
  float*       out = (float*)d_out;           // same shape
  fractal_denoise_row<<<dim3(NROWS), dim3(NTHREADS), 0, stream>>>(x, out);
}